// GraphLayer_627065225884
// MI455X (gfx1250) — compile-verified
//
#include <hip/hip_runtime.h>
#include <math.h>

typedef float v2f __attribute__((ext_vector_type(2)));
typedef float v8f __attribute__((ext_vector_type(8)));

#define N_NODES   50000
#define N_EDGES   1600000
#define IN_CH     128
#define OUT_CH    64
#define NEG_SLOPE 0.2f

// ---------------------------------------------------------------------------
// Kernel 1: xh = x @ lin_w   ([N,128] @ [128,128], fp32) via V_WMMA_F32_16X16X4_F32
// One wave computes one 16x16 output tile; 8 waves per block share 16 x-rows.
// A layout (16x4 f32): lane<16 -> K=k0,k0+1 ; lane>=16 -> K=k0+2,k0+3
// B layout (4x16 f32): v0 = B[k0 + 2*half][n0+l16], v1 = B[k0+1+2*half][n0+l16]
// D layout: VGPR r, lanes0-15 -> M=r, lanes16-31 -> M=r+8
// ---------------------------------------------------------------------------
__global__ __launch_bounds__(256) void k_gemm(const float* __restrict__ x,
                                              const float* __restrict__ w,
                                              float* __restrict__ xh)
{
    const int lane = threadIdx.x & 31;
    const int wv   = threadIdx.x >> 5;   // 0..7 -> N tile
    const int half = lane >> 4;          // 0/1
    const int l16  = lane & 15;
    const int m0   = blockIdx.x * 16;
    const int n0   = wv * 16;

    const float* arow = x + (size_t)(m0 + l16) * IN_CH + 2 * half;
    const float* bcol = w + (size_t)(2 * half) * 128 + n0 + l16;

    v8f c = {0.f, 0.f, 0.f, 0.f, 0.f, 0.f, 0.f, 0.f};
#pragma unroll
    for (int k0 = 0; k0 < IN_CH; k0 += 4) {
        float2 a2 = *(const float2*)(arow + k0);
        v2f a; a.x = a2.x; a.y = a2.y;
        v2f b; b.x = bcol[(size_t)k0 * 128];
               b.y = bcol[(size_t)(k0 + 1) * 128];
        c = __builtin_amdgcn_wmma_f32_16x16x4_f32(false, a, false, b,
                                                  (short)0, c, false, false);
    }

    float* orow = xh + (size_t)(m0 + 8 * half) * 128 + n0 + l16;
#pragma unroll
    for (int r = 0; r < 8; ++r)
        orow[(size_t)r * 128] = c[r];
}

// ---------------------------------------------------------------------------
// Kernel 2: per-node scores s_i,s_j [N,2] + initialize w with self-loop alpha.
// One wave per node; coalesced row reads; wave32 shfl_xor reduction.
// ---------------------------------------------------------------------------
__global__ __launch_bounds__(256) void k_scores(const float* __restrict__ xh,
                                                const float* __restrict__ emb,
                                                const float* __restrict__ att_i,
                                                const float* __restrict__ att_j,
                                                const float* __restrict__ att_em_i,
                                                const float* __restrict__ att_em_j,
                                                float* __restrict__ s_i,
                                                float* __restrict__ s_j,
                                                float* __restrict__ w)
{
    const int lane = threadIdx.x & 31;
    const int n    = blockIdx.x * (blockDim.x >> 5) + (threadIdx.x >> 5);

    // xh part: lane covers channels [lane*4, lane*4+3]; head = lane/16
    const int h    = lane >> 4;
    const int doff = (lane & 15) * 4;
    float4 xv = *(const float4*)(xh + (size_t)n * 128 + lane * 4);
    float4 ai = *(const float4*)(att_i + h * 64 + doff);
    float4 aj = *(const float4*)(att_j + h * 64 + doff);
    float sip = xv.x * ai.x + xv.y * ai.y + xv.z * ai.z + xv.w * ai.w;
    float sjp = xv.x * aj.x + xv.y * aj.y + xv.z * aj.z + xv.w * aj.w;

    float si0 = h ? 0.f : sip, si1 = h ? sip : 0.f;
    float sj0 = h ? 0.f : sjp, sj1 = h ? sjp : 0.f;

    // embedding part: lane covers emb channels [lane*2, lane*2+1], both heads
    float2 ev  = *(const float2*)(emb + (size_t)n * 64 + lane * 2);
    float2 ei0 = *(const float2*)(att_em_i + lane * 2);
    float2 ei1 = *(const float2*)(att_em_i + 64 + lane * 2);
    float2 ej0 = *(const float2*)(att_em_j + lane * 2);
    float2 ej1 = *(const float2*)(att_em_j + 64 + lane * 2);
    si0 += ev.x * ei0.x + ev.y * ei0.y;
    si1 += ev.x * ei1.x + ev.y * ei1.y;
    sj0 += ev.x * ej0.x + ev.y * ej0.y;
    sj1 += ev.x * ej1.x + ev.y * ej1.y;

    // wave32 reduction
#pragma unroll
    for (int off = 16; off >= 1; off >>= 1) {
        si0 += __shfl_xor(si0, off, 32);
        si1 += __shfl_xor(si1, off, 32);
        sj0 += __shfl_xor(sj0, off, 32);
        sj1 += __shfl_xor(sj1, off, 32);
    }

    if (lane == 0) {
        s_i[2 * (size_t)n]     = si0;
        s_i[2 * (size_t)n + 1] = si1;
        s_j[2 * (size_t)n]     = sj0;
        s_j[2 * (size_t)n + 1] = sj1;
        // self-loop edge (n,n): leaky_relu + softmax over heads -> init w
        float a0 = si0 + sj0; a0 = a0 > 0.f ? a0 : NEG_SLOPE * a0;
        float a1 = si1 + sj1; a1 = a1 > 0.f ? a1 : NEG_SLOPE * a1;
        float m  = fmaxf(a0, a1);
        float e0 = expf(a0 - m), e1 = expf(a1 - m);
        float inv = 1.f / (e0 + e1);
        w[2 * (size_t)n]     = e0 * inv;
        w[2 * (size_t)n + 1] = e1 * inv;
    }
}

// ---------------------------------------------------------------------------
// Kernel 3: per-edge softmaxed attention scattered into w[dst] (f32 atomics).
// ---------------------------------------------------------------------------
__global__ __launch_bounds__(256) void k_edges(const int* __restrict__ esrc,
                                               const int* __restrict__ edst,
                                               const float* __restrict__ s_i,
                                               const float* __restrict__ s_j,
                                               float* __restrict__ w)
{
    const int e = blockIdx.x * blockDim.x + threadIdx.x;   // grid covers exactly N_EDGES
    const int s = esrc[e];
    const int d = edst[e];
    float2 ai = *(const float2*)(s_i + 2 * (size_t)s);
    float2 aj = *(const float2*)(s_j + 2 * (size_t)d);
    float a0 = ai.x + aj.x; a0 = a0 > 0.f ? a0 : NEG_SLOPE * a0;
    float a1 = ai.y + aj.y; a1 = a1 > 0.f ? a1 : NEG_SLOPE * a1;
    float m  = fmaxf(a0, a1);
    float e0 = expf(a0 - m), e1 = expf(a1 - m);
    float inv = 1.f / (e0 + e1);
    unsafeAtomicAdd(w + 2 * (size_t)d,     e0 * inv);
    unsafeAtomicAdd(w + 2 * (size_t)d + 1, e1 * inv);
}

// ---------------------------------------------------------------------------
// Kernel 4: out[n,d] = 0.5 * (xh[n,0,d]*w[n,0] + xh[n,1,d]*w[n,1])
// ---------------------------------------------------------------------------
__global__ __launch_bounds__(256) void k_out(const float* __restrict__ xh,
                                             const float* __restrict__ w,
                                             float* __restrict__ out)
{
    const int idx = blockIdx.x * blockDim.x + threadIdx.x;  // n*64 + d
    const int n = idx >> 6;
    const int d = idx & 63;
    float w0 = w[2 * (size_t)n];
    float w1 = w[2 * (size_t)n + 1];
    out[idx] = 0.5f * (xh[(size_t)n * 128 + d]      * w0 +
                       xh[(size_t)n * 128 + 64 + d] * w1);
}

extern "C" void kernel_launch(void* const* d_in, const int* in_sizes, int n_in,
                              void* d_out, int out_size, void* d_ws, size_t ws_size,
                              hipStream_t stream)
{
    const float* x         = (const float*)d_in[0];
    const float* emb       = (const float*)d_in[1];
    const float* lin_w     = (const float*)d_in[2];
    const float* att_i     = (const float*)d_in[3];
    const float* att_j     = (const float*)d_in[4];
    const float* att_em_i  = (const float*)d_in[5];
    const float* att_em_j  = (const float*)d_in[6];
    const int*   esrc      = (const int*)d_in[7];
    const int*   edst      = (const int*)d_in[8];
    float*       out       = (float*)d_out;

    float* ws  = (float*)d_ws;
    float* xh  = ws;                                  // N*128 floats (25.6 MB)
    float* s_i = xh  + (size_t)N_NODES * 128;         // N*2
    float* s_j = s_i + 2 * (size_t)N_NODES;           // N*2
    float* wsm = s_j + 2 * (size_t)N_NODES;           // N*2

    k_gemm  <<<N_NODES / 16, 256, 0, stream>>>(x, lin_w, xh);
    k_scores<<<N_NODES / 8,  256, 0, stream>>>(xh, emb, att_i, att_j,
                                               att_em_i, att_em_j, s_i, s_j, wsm);
    k_edges <<<N_EDGES / 256, 256, 0, stream>>>(esrc, edst, s_i, s_j, wsm);
    k_out   <<<(N_NODES * OUT_CH) / 256, 256, 0, stream>>>(xh, wsm, out);
}